// MDCN_73108933312964
// MI455X (gfx1250) — compile-verified
//
#include <hip/hip_runtime.h>

#define FEAT   256
#define MIX    16
#define BATCH  16384
#define HID    16
#define NMID   16
#define BN_EPS 1e-5f
#define RHO_1  0.95f
#define TAU_INV 1e-4f

typedef __attribute__((ext_vector_type(2))) float v2f;
typedef __attribute__((ext_vector_type(4))) float f4v;
typedef __attribute__((ext_vector_type(8))) float v8f;

// D = A(16x4) * B(4x16) + C, f32, wave32.
__device__ __forceinline__ v8f wmma4(v2f a, v2f b, v8f c) {
  return __builtin_amdgcn_wmma_f32_16x16x4_f32(false, a, false, b, (short)0, c,
                                               false, false);
}

__device__ __forceinline__ float dot4(f4v a, f4v b) {
  return a[0]*b[0] + a[1]*b[1] + a[2]*b[2] + a[3]*b[3];
}

// ---------------- workspace layout (float offsets) ----------------
// rawA, rawB: ping-pong hidden activations (B x HID)
// fraw, fbn : (B x FEAT)
// prePi, preVar : (B x MIX)
// stats region (zeroed each call, 2048 floats):
//   layer L in [0,16]: +32*L  (sum[16], sumsq[16])
//   f stats: +544 (sum[256], sumsq[256])
//   pi stats: +1056, var stats: +1088
// sdZ: +2048 past stats (256 floats, written not zeroed)
static const size_t OFF_RAWA   = 0;
static const size_t OFF_RAWB   = (size_t)BATCH * HID;
static const size_t OFF_FRAW   = OFF_RAWB + (size_t)BATCH * HID;
static const size_t OFF_FBN    = OFF_FRAW + (size_t)BATCH * FEAT;
static const size_t OFF_PREPI  = OFF_FBN + (size_t)BATCH * FEAT;
static const size_t OFF_PREVAR = OFF_PREPI + (size_t)BATCH * MIX;
static const size_t OFF_STATS  = OFF_PREVAR + (size_t)BATCH * MIX;
static const size_t OFF_SDZ    = OFF_STATS + 2048;

// ---------------- init: zero stats, precompute sdZ = exp(0.5*logvarZ) -------
__global__ void k_init(float* __restrict__ stats, const float* __restrict__ logvarZ) {
  int t = threadIdx.x;                       // 1 block x 256
  for (int i = t; i < 2048; i += 256) stats[i] = 0.f;
  stats[2048 + t] = expf(0.5f * logvarZ[t]);
}

// ---------------- start layer: relu(feature @ Ws^T + bs) --------------------
__global__ void k_start(const float* __restrict__ feat, const float* __restrict__ Ws,
                        const float* __restrict__ bs, float* __restrict__ rawOut,
                        float* __restrict__ statsOut) {
  __shared__ float sred[32];
  int t = threadIdx.x;
  if (t < 32) sred[t] = 0.f;
  __syncthreads();
  int wave = (blockIdx.x * blockDim.x + t) >> 5;
  int lane = t & 31, r = lane & 15, hi = lane >> 4;
  int row0 = wave * 16;
  v8f acc = {};
  for (int kk = 0; kk < FEAT; kk += 4) {
    int c0 = kk + 2 * hi;
    v2f a = *(const v2f*)(feat + (size_t)(row0 + r) * FEAT + c0);
    v2f b = *(const v2f*)(Ws + (size_t)r * FEAT + c0);
    acc = wmma4(a, b, acc);
  }
  float bias = bs[r];
  float ls = 0.f, lq = 0.f;
  for (int v = 0; v < 8; ++v) {
    float y = fmaxf(acc[v] + bias, 0.f);
    rawOut[(size_t)(row0 + v + 8 * hi) * HID + r] = y;
    ls += y; lq += y * y;
  }
  atomicAdd(&sred[r], ls);
  atomicAdd(&sred[16 + r], lq);
  __syncthreads();
  if (t < 32) atomicAdd(&statsOut[t], sred[t]);
}

// ---------------- mid layer: relu(BN(rawIn) @ Wm^T + bm) --------------------
__global__ void k_mid(const float* __restrict__ rawIn, const float* __restrict__ statsIn,
                      const float* __restrict__ Wm, const float* __restrict__ bm,
                      const float* __restrict__ gIn, const float* __restrict__ beIn,
                      float* __restrict__ rawOut, float* __restrict__ statsOut) {
  __shared__ float sscale[HID], sshift[HID], sred[32];
  int t = threadIdx.x;
  if (t < HID) {
    float m = statsIn[t] * (1.f / BATCH);
    float v = statsIn[HID + t] * (1.f / BATCH) - m * m;
    float sc = gIn[t] * rsqrtf(v + BN_EPS);
    sscale[t] = sc;
    sshift[t] = beIn[t] - m * sc;
  }
  if (t < 32) sred[t] = 0.f;
  __syncthreads();
  int wave = (blockIdx.x * blockDim.x + t) >> 5;
  int lane = t & 31, r = lane & 15, hi = lane >> 4;
  int row0 = wave * 16;
  v8f acc = {};
  for (int kk = 0; kk < HID; kk += 4) {
    int c0 = kk + 2 * hi;
    v2f x = *(const v2f*)(rawIn + (size_t)(row0 + r) * HID + c0);
    v2f a;
    a[0] = x[0] * sscale[c0] + sshift[c0];
    a[1] = x[1] * sscale[c0 + 1] + sshift[c0 + 1];
    v2f b = *(const v2f*)(Wm + (size_t)r * HID + c0);
    acc = wmma4(a, b, acc);
  }
  float bias = bm[r];
  float ls = 0.f, lq = 0.f;
  for (int v = 0; v < 8; ++v) {
    float y = fmaxf(acc[v] + bias, 0.f);
    rawOut[(size_t)(row0 + v + 8 * hi) * HID + r] = y;
    ls += y; lq += y * y;
  }
  atomicAdd(&sred[r], ls);
  atomicAdd(&sred[16 + r], lq);
  __syncthreads();
  if (t < 32) atomicAdd(&statsOut[t], sred[t]);
}

// ---------------- end layer: relu(BN(rawIn) @ We^T + be) -> f_raw (B x 256) -
__global__ void k_end(const float* __restrict__ rawIn, const float* __restrict__ statsIn,
                      const float* __restrict__ gIn, const float* __restrict__ beIn,
                      const float* __restrict__ We, const float* __restrict__ bE,
                      float* __restrict__ fraw, float* __restrict__ statsF) {
  __shared__ float sscale[HID], sshift[HID];
  __shared__ float sF[512];
  int t = threadIdx.x;
  if (t < HID) {
    float m = statsIn[t] * (1.f / BATCH);
    float v = statsIn[HID + t] * (1.f / BATCH) - m * m;
    float sc = gIn[t] * rsqrtf(v + BN_EPS);
    sscale[t] = sc;
    sshift[t] = beIn[t] - m * sc;
  }
  for (int i = t; i < 512; i += blockDim.x) sF[i] = 0.f;
  __syncthreads();
  int wave = (blockIdx.x * blockDim.x + t) >> 5;
  int lane = t & 31, r = lane & 15, hi = lane >> 4;
  int row0 = wave * 16;
  v2f a[4];
  for (int kk4 = 0; kk4 < 4; ++kk4) {
    int c0 = kk4 * 4 + 2 * hi;
    v2f x = *(const v2f*)(rawIn + (size_t)(row0 + r) * HID + c0);
    a[kk4][0] = x[0] * sscale[c0] + sshift[c0];
    a[kk4][1] = x[1] * sscale[c0 + 1] + sshift[c0 + 1];
  }
  for (int nt = 0; nt < 16; ++nt) {
    int n = nt * 16 + r;
    v8f acc = {};
    for (int kk4 = 0; kk4 < 4; ++kk4) {
      int c0 = kk4 * 4 + 2 * hi;
      v2f b = *(const v2f*)(We + (size_t)n * HID + c0);
      acc = wmma4(a[kk4], b, acc);
    }
    float bias = bE[n];
    float ls = 0.f, lq = 0.f;
    for (int v = 0; v < 8; ++v) {
      float y = fmaxf(acc[v] + bias, 0.f);
      fraw[(size_t)(row0 + v + 8 * hi) * FEAT + n] = y;
      ls += y; lq += y * y;
    }
    atomicAdd(&sF[n], ls);
    atomicAdd(&sF[256 + n], lq);
  }
  __syncthreads();
  for (int i = t; i < 512; i += blockDim.x) atomicAdd(&statsF[i], sF[i]);
}

// ---------------- f = BN(f_raw) (elementwise, vectorized) -------------------
__global__ void k_bnf(const float* __restrict__ fraw, const float* __restrict__ statsF,
                      const float* __restrict__ g, const float* __restrict__ be,
                      float* __restrict__ fbn) {
  __shared__ float sc[256], sh[256];
  int t = threadIdx.x;  // blockDim == 256
  {
    float m = statsF[t] * (1.f / BATCH);
    float v = statsF[256 + t] * (1.f / BATCH) - m * m;
    float s = g[t] * rsqrtf(v + BN_EPS);
    sc[t] = s;
    sh[t] = be[t] - m * s;
  }
  __syncthreads();
  size_t total4 = (size_t)BATCH * FEAT / 4;
  size_t stride = (size_t)gridDim.x * blockDim.x;
  for (size_t i = (size_t)blockIdx.x * blockDim.x + t; i < total4; i += stride) {
    int c0 = (int)((i * 4) & (FEAT - 1));
    f4v x = *(const f4v*)(fraw + i * 4);
    f4v y;
    y[0] = x[0] * sc[c0]     + sh[c0];
    y[1] = x[1] * sc[c0 + 1] + sh[c0 + 1];
    y[2] = x[2] * sc[c0 + 2] + sh[c0 + 2];
    y[3] = x[3] * sc[c0 + 3] + sh[c0 + 3];
    *(f4v*)(fbn + i * 4) = y;
  }
}

// ---------------- heads: rho (final), pi/var pre-activations + stats --------
__global__ void k_heads1(const float* __restrict__ fbn,
                         const float* __restrict__ Wr, const float* __restrict__ brh,
                         const float* __restrict__ Wp, const float* __restrict__ bph,
                         const float* __restrict__ Wv, const float* __restrict__ bvh,
                         float* __restrict__ outRho, float* __restrict__ prePi,
                         float* __restrict__ preVar, float* __restrict__ statsPi,
                         float* __restrict__ statsVar) {
  __shared__ float sred[64];
  int t = threadIdx.x;
  if (t < 64) sred[t] = 0.f;
  __syncthreads();
  int wave = (blockIdx.x * blockDim.x + t) >> 5;
  int lane = t & 31, r = lane & 15, hi = lane >> 4;
  int row0 = wave * 16;
  v8f ar = {}, ap = {}, av = {};
  for (int kk = 0; kk < FEAT; kk += 4) {
    int c0 = kk + 2 * hi;
    v2f a  = *(const v2f*)(fbn + (size_t)(row0 + r) * FEAT + c0);
    v2f b0 = *(const v2f*)(Wr + (size_t)r * FEAT + c0);
    v2f b1 = *(const v2f*)(Wp + (size_t)r * FEAT + c0);
    v2f b2 = *(const v2f*)(Wv + (size_t)r * FEAT + c0);
    ar = wmma4(a, b0, ar);
    ap = wmma4(a, b1, ap);
    av = wmma4(a, b2, av);
  }
  float biasR = brh[r], biasP = bph[r], biasV = bvh[r];
  float sp = 0.f, sp2 = 0.f, sv = 0.f, sv2 = 0.f;
  for (int v = 0; v < 8; ++v) {
    size_t row = (size_t)(row0 + v + 8 * hi);
    float yr = tanhf(ar[v] + biasR);
    if (r == 0) yr = RHO_1;
    outRho[row * MIX + r] = yr;
    float yp = ap[v] + biasP;
    prePi[row * MIX + r] = yp; sp += yp; sp2 += yp * yp;
    float yv = av[v] + biasV;
    preVar[row * MIX + r] = yv; sv += yv; sv2 += yv * yv;
  }
  atomicAdd(&sred[r], sp);
  atomicAdd(&sred[16 + r], sp2);
  atomicAdd(&sred[32 + r], sv);
  atomicAdd(&sred[48 + r], sv2);
  __syncthreads();
  if (t < 32) atomicAdd(&statsPi[t], sred[t]);
  else if (t < 64) atomicAdd(&statsVar[t - 32], sred[t]);
}

// ---------------- heads2: softmax(relu(BN(pi))), var head -------------------
__global__ void k_heads2(const float* __restrict__ prePi, const float* __restrict__ preVar,
                         const float* __restrict__ statsPi, const float* __restrict__ statsVar,
                         const float* __restrict__ g_pi, const float* __restrict__ be_pi,
                         const float* __restrict__ g_var, const float* __restrict__ be_var,
                         const float* __restrict__ outRho,
                         float* __restrict__ outPi, float* __restrict__ outVar) {
  __shared__ float scp[16], shp[16], scv[16], shv[16];
  int t = threadIdx.x;
  if (t < 16) {
    float m = statsPi[t] * (1.f / BATCH);
    float v = statsPi[16 + t] * (1.f / BATCH) - m * m;
    float s = g_pi[t] * rsqrtf(v + BN_EPS);
    scp[t] = s; shp[t] = be_pi[t] - m * s;
  } else if (t < 32) {
    int j = t - 16;
    float m = statsVar[j] * (1.f / BATCH);
    float v = statsVar[16 + j] * (1.f / BATCH) - m * m;
    float s = g_var[j] * rsqrtf(v + BN_EPS);
    scv[j] = s; shv[j] = be_var[j] - m * s;
  }
  __syncthreads();
  size_t b = (size_t)blockIdx.x * blockDim.x + t;
  float tv[16];
  float mx = -1e30f;
  for (int j = 0; j < 16; ++j) {
    float x = fmaxf(prePi[b * MIX + j] * scp[j] + shp[j], 0.f);
    tv[j] = x;
    mx = fmaxf(mx, x);
  }
  float s = 0.f;
  for (int j = 0; j < 16; ++j) { float e = expf(tv[j] - mx); tv[j] = e; s += e; }
  float inv = 1.f / s;
  for (int j = 0; j < 16; ++j) outPi[b * MIX + j] = tv[j] * inv;
  for (int j = 0; j < 16; ++j) {
    float v0 = expf(fmaxf(preVar[b * MIX + j] * scv[j] + shv[j], 0.f));
    float rho = outRho[b * MIX + j];
    outVar[b * MIX + j] = (1.f - rho * rho) * v0 + TAU_INV;
  }
}

// ---------------- mu: stream W_noise/Z_noise once, one wave per batch row ---
__global__ void k_mu(const float* __restrict__ fbn, const float* __restrict__ Wn,
                     const float* __restrict__ Zn, const float* __restrict__ muW,
                     const float* __restrict__ sdZ, const float* __restrict__ outRho,
                     float* __restrict__ outMu) {
  int wave = (blockIdx.x * blockDim.x + threadIdx.x) >> 5;
  int lane = threadIdx.x & 31;
  size_t b = (size_t)wave;
  const f4v* f4 = (const f4v*)(fbn + b * FEAT);
  const f4v* s4 = (const f4v*)sdZ;
  const f4v* m4 = (const f4v*)muW;
  f4v fA = f4[lane],      fB = f4[32 + lane];
  f4v sA = s4[lane],      sB = s4[32 + lane];
  f4v mA = m4[lane],      mB = m4[32 + lane];
  f4v fsA = fA * sA, fsB = fB * sB;
  float dmu = dot4(mA, fA) + dot4(mB, fB);
#pragma unroll
  for (int off = 16; off; off >>= 1) dmu += __shfl_xor(dmu, off, 32);
  for (int m = 0; m < MIX; ++m) {
    const f4v* w4 = (const f4v*)(Wn + ((size_t)m * BATCH + b) * FEAT);
    const f4v* z4 = (const f4v*)(Zn + ((size_t)m * BATCH + b) * FEAT);
    f4v wA = __builtin_nontemporal_load(w4 + lane);
    f4v wB = __builtin_nontemporal_load(w4 + 32 + lane);
    f4v zA = __builtin_nontemporal_load(z4 + lane);
    f4v zB = __builtin_nontemporal_load(z4 + 32 + lane);
    float aw = dot4(wA, fsA) + dot4(wB, fsB);
    float az = dot4(zA, fsA) + dot4(zB, fsB);
#pragma unroll
    for (int off = 16; off; off >>= 1) {
      aw += __shfl_xor(aw, off, 32);
      az += __shfl_xor(az, off, 32);
    }
    if (lane == 0) {
      float rho = outRho[b * MIX + m];
      float a2 = 1.f - rho * rho;
      outMu[b * MIX + m] = rho * dmu + a2 * (rho * aw + a2 * az);
    }
  }
}

extern "C" void kernel_launch(void* const* d_in, const int* in_sizes, int n_in,
                              void* d_out, int out_size, void* d_ws, size_t ws_size,
                              hipStream_t stream) {
  (void)in_sizes; (void)n_in; (void)out_size; (void)ws_size;
  const float* feature  = (const float*)d_in[0];
  const float* Wn       = (const float*)d_in[1];
  const float* Zn       = (const float*)d_in[2];
  const float* W_start  = (const float*)d_in[3];
  const float* b_start  = (const float*)d_in[4];
  const float* g_start  = (const float*)d_in[5];
  const float* be_start = (const float*)d_in[6];
  const float* W_mid    = (const float*)d_in[7];
  const float* b_mid    = (const float*)d_in[8];
  const float* g_mid    = (const float*)d_in[9];
  const float* be_mid   = (const float*)d_in[10];
  const float* W_end    = (const float*)d_in[11];
  const float* b_end    = (const float*)d_in[12];
  const float* g_end    = (const float*)d_in[13];
  const float* be_end   = (const float*)d_in[14];
  const float* Wr       = (const float*)d_in[15];
  const float* br       = (const float*)d_in[16];
  const float* Wp       = (const float*)d_in[17];
  const float* bp       = (const float*)d_in[18];
  const float* g_pi     = (const float*)d_in[19];
  const float* be_pi    = (const float*)d_in[20];
  const float* Wv       = (const float*)d_in[21];
  const float* bv       = (const float*)d_in[22];
  const float* g_var    = (const float*)d_in[23];
  const float* be_var   = (const float*)d_in[24];
  const float* muW      = (const float*)d_in[25];
  // d_in[26] = logvarW: cancels algebraically, unused
  const float* logvarZ  = (const float*)d_in[27];

  float* ws = (float*)d_ws;
  float* rawA   = ws + OFF_RAWA;
  float* rawB   = ws + OFF_RAWB;
  float* fraw   = ws + OFF_FRAW;
  float* fbn    = ws + OFF_FBN;
  float* prePi  = ws + OFF_PREPI;
  float* preVar = ws + OFF_PREVAR;
  float* stats  = ws + OFF_STATS;
  float* sdZ    = ws + OFF_SDZ;
  float* statsF   = stats + 544;
  float* statsPi  = stats + 1056;
  float* statsVar = stats + 1088;

  float* out    = (float*)d_out;
  float* outRho = out;
  float* outPi  = out + (size_t)BATCH * MIX;
  float* outMu  = out + 2 * (size_t)BATCH * MIX;
  float* outVar = out + 3 * (size_t)BATCH * MIX;

  const int TILE_BLOCKS = (BATCH / 16) / 8;  // 128: one wave per 16-row tile

  k_init<<<1, 256, 0, stream>>>(stats, logvarZ);
  k_start<<<TILE_BLOCKS, 256, 0, stream>>>(feature, W_start, b_start, rawA, stats);
  float* cur = rawA;
  float* nxt = rawB;
  for (int i = 0; i < NMID; ++i) {
    const float* gI  = (i == 0) ? g_start : g_mid;
    const float* beI = (i == 0) ? be_start : be_mid;
    k_mid<<<TILE_BLOCKS, 256, 0, stream>>>(cur, stats + 32 * i, W_mid, b_mid,
                                           gI, beI, nxt, stats + 32 * (i + 1));
    float* tmp = cur; cur = nxt; nxt = tmp;
  }
  k_end<<<TILE_BLOCKS, 256, 0, stream>>>(cur, stats + 32 * NMID, g_mid, be_mid,
                                         W_end, b_end, fraw, statsF);
  k_bnf<<<1024, 256, 0, stream>>>(fraw, statsF, g_end, be_end, fbn);
  k_heads1<<<TILE_BLOCKS, 256, 0, stream>>>(fbn, Wr, br, Wp, bp, Wv, bv,
                                            outRho, prePi, preVar, statsPi, statsVar);
  k_heads2<<<BATCH / 256, 256, 0, stream>>>(prePi, preVar, statsPi, statsVar,
                                            g_pi, be_pi, g_var, be_var,
                                            outRho, outPi, outVar);
  k_mu<<<BATCH / 8, 256, 0, stream>>>(fbn, Wn, Zn, muW, sdZ, outRho, outMu);
}